// MultiheadAttention_6957847020075
// MI455X (gfx1250) — compile-verified
//
#include <hip/hip_runtime.h>

typedef __attribute__((ext_vector_type(2)))  float v2f;
typedef __attribute__((ext_vector_type(8)))  float v8f;
typedef __attribute__((ext_vector_type(16))) float v16f;

#define D_DIM 1024
#define NHEAD 16
#define HDIM  64
#define BATCH 2
#define SEQ   2048

static __device__ __forceinline__ v8f wmma4(v2f a, v2f b, v8f c) {
  // V_WMMA_F32_16X16X4_F32: D = A(16x4,f32) x B(4x16,f32) + C(16x16,f32)
  // args: (neg_a, A, neg_b, B, c_mod, C, reuse_a, reuse_b)
  return __builtin_amdgcn_wmma_f32_16x16x4_f32(false, a, false, b, (short)0, c,
                                               false, false);
}

// ---------------------------------------------------------------------------
// Generic fp32 GEMM + bias: C[M,N] = A[M,K] @ B[K,N] + bias[N]
// A,B row-major. Block = 256 threads (8 waves). Wave tile = 32x32 (2x2 WMMA
// accumulators). Block tile = 128(M) x 64(N). M%128==0, N%64==0, K%4==0.
//
// Fragment layout (32-bit WMMA, wave32):
//   A 16x4 : lane (half=lane/16, lm=lane%16): vgpr0=A[lm][2*half], vgpr1=+1
//   B 4x16 : vgpr0 = B[2*half][lm], vgpr1 = B[2*half+1][lm]
//   C 16x16: vgpr r  = C[r + 8*half][lm]
// ---------------------------------------------------------------------------
__global__ __launch_bounds__(256)
void gemm_bias_kernel(const float* __restrict__ A, const float* __restrict__ B,
                      const float* __restrict__ bias, float* __restrict__ C,
                      int M, int N, int K) {
  const int lane = threadIdx.x & 31;
  const int wave = threadIdx.x >> 5;
  const int half = lane >> 4;
  const int lm   = lane & 15;
  const int wy   = wave >> 1;                 // 0..3
  const int wx   = wave & 1;                  // 0..1
  const int mbase = blockIdx.y * 128 + wy * 32;
  const int nbase = blockIdx.x * 64  + wx * 32;

  v8f acc00 = {}, acc01 = {}, acc10 = {}, acc11 = {};

  const float* a0p = A + (size_t)(mbase + lm)      * K + 2 * half;
  const float* a1p = A + (size_t)(mbase + 16 + lm) * K + 2 * half;
  const float* b0p = B + (size_t)(2 * half) * N + nbase + lm;   // col tile 0
  const float* b1p = b0p + 16;                                  // col tile 1

  for (int k = 0; k < K; k += 4) {
    v2f a0 = *(const v2f*)(a0p + k);
    v2f a1 = *(const v2f*)(a1p + k);
    v2f b0, b1;
    b0.x = b0p[(size_t)k * N];
    b0.y = b0p[(size_t)(k + 1) * N];
    b1.x = b1p[(size_t)k * N];
    b1.y = b1p[(size_t)(k + 1) * N];
    acc00 = wmma4(a0, b0, acc00);
    acc01 = wmma4(a0, b1, acc01);
    acc10 = wmma4(a1, b0, acc10);
    acc11 = wmma4(a1, b1, acc11);
  }

  const int col0 = nbase + lm;
  const int col1 = nbase + 16 + lm;
  const float bs0 = bias[col0];
  const float bs1 = bias[col1];
#pragma unroll
  for (int r = 0; r < 8; ++r) {
    const int row0 = mbase + r + 8 * half;
    const int row1 = row0 + 16;
    C[(size_t)row0 * N + col0] = acc00[r] + bs0;
    C[(size_t)row0 * N + col1] = acc01[r] + bs1;
    C[(size_t)row1 * N + col0] = acc10[r] + bs0;
    C[(size_t)row1 * N + col1] = acc11[r] + bs1;
  }
}

// ---------------------------------------------------------------------------
// Fused scores + softmax-over-heads.
//   scores[b,l,n,h] = 0.125 * sum_d q[b,l,d*16+h] * k[b,n,d*16+h]
//   probs[b,l,n,:]  = softmax_h(scores[b,l,n,:])
// One wave handles a 16x16 (l,n) tile for ALL 16 heads (16 v8f accumulators).
// The 16 heads of one K-element are 16 consecutive floats -> one v16f load
// feeds all heads. Softmax over h is lane-local (all 16 scores of an (l,n)
// element sit in the same register slot across the 16 accumulators).
// Block = 8 waves = 2 l-tiles x 4 n-tiles (covers l:32, n:64).
// Grid = (SEQ/64, SEQ/32, BATCH).
// ---------------------------------------------------------------------------
__global__ __launch_bounds__(256)
void scores_softmax_kernel(const float* __restrict__ q,
                           const float* __restrict__ kmat,
                           float* __restrict__ probs) {
  const int lane = threadIdx.x & 31;
  const int wave = threadIdx.x >> 5;
  const int half = lane >> 4;
  const int lm   = lane & 15;
  const int wl   = wave >> 2;                 // 0..1
  const int wn   = wave & 3;                  // 0..3
  const int b    = blockIdx.z;
  const int lbase = blockIdx.y * 32 + wl * 16;
  const int nbase = blockIdx.x * 64 + wn * 16;

  const float* qrow = q    + ((size_t)b * SEQ + lbase + lm) * D_DIM;
  const float* krow = kmat + ((size_t)b * SEQ + nbase + lm) * D_DIM;

  v8f acc[NHEAD] = {};

  for (int d0 = 0; d0 < HDIM; d0 += 4) {
    const int dA = d0 + 2 * half;             // lane-half's first K element
    // 16 heads x 2 consecutive K elements, contiguous 128B per matrix
    v16f q0 = *(const v16f*)(qrow + (size_t)dA * NHEAD);
    v16f q1 = *(const v16f*)(qrow + (size_t)(dA + 1) * NHEAD);
    v16f k0 = *(const v16f*)(krow + (size_t)dA * NHEAD);
    v16f k1 = *(const v16f*)(krow + (size_t)(dA + 1) * NHEAD);
#pragma unroll
    for (int h = 0; h < NHEAD; ++h) {
      v2f a; a.x = q0[h]; a.y = q1[h];
      v2f bb; bb.x = k0[h]; bb.y = k1[h];
      acc[h] = wmma4(a, bb, acc[h]);
    }
  }

  // Softmax over heads, entirely in registers; coalesced 64B/lane stores.
#pragma unroll
  for (int r = 0; r < 8; ++r) {
    float e[NHEAD];
    float m = -3.4e38f;
#pragma unroll
    for (int h = 0; h < NHEAD; ++h) {
      const float s = 0.125f * acc[h][r];     // 1/sqrt(HDIM)
      e[h] = s;
      m = fmaxf(m, s);
    }
    float sum = 0.f;
#pragma unroll
    for (int h = 0; h < NHEAD; ++h) { e[h] = __expf(e[h] - m); sum += e[h]; }
    const float inv = 1.0f / sum;

    const int l = lbase + r + 8 * half;
    const int n = nbase + lm;
    v16f pv;
#pragma unroll
    for (int h = 0; h < NHEAD; ++h) pv[h] = e[h] * inv;
    *(v16f*)(probs + (((size_t)b * SEQ + l) * SEQ + n) * NHEAD) = pv;
  }
}

// ---------------------------------------------------------------------------
// P @ V per head: out[b,l,d*16+h] = sum_n probs[b,l,n,h] * v[b,n,d*16+h]
// One wave: 16(l) x 16(d) tile for ALL 16 heads; contiguous v16f reads of
// probs and v; contiguous 64B/lane stores of out.
// Block = 8 waves = 2 l-tiles x 4 d-tiles (covers l:32, full D).
// Grid = (1, SEQ/32, BATCH).
// ---------------------------------------------------------------------------
__global__ __launch_bounds__(256)
void pv_kernel(const float* __restrict__ probs, const float* __restrict__ vmat,
               float* __restrict__ out) {
  const int lane = threadIdx.x & 31;
  const int wave = threadIdx.x >> 5;
  const int half = lane >> 4;
  const int lm   = lane & 15;
  const int wl   = wave >> 2;                 // 0..1
  const int wd   = wave & 3;                  // 0..3
  const int b    = blockIdx.z;
  const int lbase = blockIdx.y * 32 + wl * 16;
  const int dbase = wd * 16;

  const float* prow = probs + (((size_t)b * SEQ + lbase + lm) * SEQ) * NHEAD;
  const float* vcol = vmat + (size_t)b * SEQ * D_DIM
                           + (size_t)(dbase + lm) * NHEAD;

  v8f acc[NHEAD] = {};

  for (int n0 = 0; n0 < SEQ; n0 += 4) {
    const int nA = n0 + 2 * half;
    v16f p0 = *(const v16f*)(prow + (size_t)nA * NHEAD);
    v16f p1 = *(const v16f*)(prow + (size_t)(nA + 1) * NHEAD);
    v16f v0 = *(const v16f*)(vcol + (size_t)nA * D_DIM);
    v16f v1 = *(const v16f*)(vcol + (size_t)(nA + 1) * D_DIM);
#pragma unroll
    for (int h = 0; h < NHEAD; ++h) {
      v2f a; a.x = p0[h]; a.y = p1[h];
      v2f bb; bb.x = v0[h]; bb.y = v1[h];
      acc[h] = wmma4(a, bb, acc[h]);
    }
  }

#pragma unroll
  for (int r = 0; r < 8; ++r) {
    const int l = lbase + r + 8 * half;
    v16f o;
#pragma unroll
    for (int h = 0; h < NHEAD; ++h) o[h] = acc[h][r];
    *(v16f*)(out + ((size_t)b * SEQ + l) * D_DIM + (size_t)(dbase + lm) * NHEAD) = o;
  }
}

// ---------------------------------------------------------------------------
extern "C" void kernel_launch(void* const* d_in, const int* in_sizes, int n_in,
                              void* d_out, int out_size, void* d_ws, size_t ws_size,
                              hipStream_t stream) {
  (void)in_sizes; (void)n_in; (void)out_size; (void)ws_size;

  const float* query = (const float*)d_in[0];
  const float* key_  = (const float*)d_in[1];
  const float* value = (const float*)d_in[2];
  const float* Wq    = (const float*)d_in[3];
  const float* bq    = (const float*)d_in[4];
  const float* Wk    = (const float*)d_in[5];
  const float* bk    = (const float*)d_in[6];
  const float* Wv    = (const float*)d_in[7];
  const float* bv    = (const float*)d_in[8];
  const float* Wo    = (const float*)d_in[9];
  const float* bo    = (const float*)d_in[10];

  const size_t act = (size_t)BATCH * SEQ * D_DIM;   // 4,194,304 floats

  float* out   = (float*)d_out;                     // [B,S,D]
  float* probs = out + act;                         // [B,S,S,NH]

  float* qp = (float*)d_ws;                         // 3 x 16.8 MB workspace
  float* kp = qp + act;
  float* vp = kp + act;
  float* ao = qp;   // attention output aliases q (q dead after score kernel)

  const dim3 blk(256);
  const dim3 gProj(D_DIM / 64, (BATCH * SEQ) / 128);        // (16, 32)
  const dim3 gScore(SEQ / 64, SEQ / 32, BATCH);             // (32, 64, 2)
  const dim3 gPV(1, SEQ / 32, BATCH);                       // (1, 64, 2)

  const int M = BATCH * SEQ;

  gemm_bias_kernel<<<gProj, blk, 0, stream>>>(query, Wq, bq, qp, M, D_DIM, D_DIM);
  gemm_bias_kernel<<<gProj, blk, 0, stream>>>(key_,  Wk, bk, kp, M, D_DIM, D_DIM);
  gemm_bias_kernel<<<gProj, blk, 0, stream>>>(value, Wv, bv, vp, M, D_DIM, D_DIM);

  scores_softmax_kernel<<<gScore, blk, 0, stream>>>(qp, kp, probs);

  pv_kernel<<<gPV, blk, 0, stream>>>(probs, vp, ao);

  gemm_bias_kernel<<<gProj, blk, 0, stream>>>(ao, Wo, bo, out, M, D_DIM, D_DIM);
}